// MoeMatmul_4861902979822
// MI455X (gfx1250) — compile-verified
//
#include <hip/hip_runtime.h>

typedef __attribute__((ext_vector_type(2))) float v2f;
typedef __attribute__((ext_vector_type(8))) float v8f;
typedef int v4i __attribute__((vector_size(16)));   // matches builtin param type

#define TOKENS  4096
#define IN_DIM  4096
#define OUT_DIM 14336

#define BM 128
#define BN 128
#define BK 16
#define LDS_STRIDE 20   // 16 K-floats + 4 pad: 16B-aligned rows, conflict-free frag reads

// ---- CDNA5 async global->LDS path ------------------------------------------
#if defined(__has_builtin)
#  if __has_builtin(__builtin_amdgcn_global_load_async_to_lds_b128)
#    define HAVE_ASYNC_LDS 1
#  endif
#  if __has_builtin(__builtin_amdgcn_s_wait_asynccnt)
#    define HAVE_WAIT_ASYNC_BUILTIN 1
#  endif
#endif
#ifndef HAVE_ASYNC_LDS
#  define HAVE_ASYNC_LDS 0
#endif
#ifndef HAVE_WAIT_ASYNC_BUILTIN
#  define HAVE_WAIT_ASYNC_BUILTIN 0
#endif

template <int N>
__device__ __forceinline__ void wait_async() {
#if HAVE_WAIT_ASYNC_BUILTIN
    __builtin_amdgcn_s_wait_asynccnt(N);
#else
    asm volatile("s_wait_asynccnt %0" ::"i"(N) : "memory");
#endif
}

__device__ __forceinline__ void async_cp_b128(const float* gsrc, float* ldst) {
#if HAVE_ASYNC_LDS
    __builtin_amdgcn_global_load_async_to_lds_b128(
        (v4i*)gsrc, (v4i*)ldst, /*imm offset*/ 0, /*cpol*/ 0);
#else
    *(float4*)ldst = *(const float4*)gsrc;   // sync fallback
#endif
}

__global__ __launch_bounds__(256, 2)
void moe_gemm_f32_wmma(const float* __restrict__ state,
                       const float* __restrict__ w,
                       const int*   __restrict__ expert_id,
                       float* __restrict__ out)
{
    __shared__ __align__(16) float As[2][BM * LDS_STRIDE];
    __shared__ __align__(16) float Bs[2][BN * LDS_STRIDE];

    const int tid   = threadIdx.x;
    const int lane  = tid & 31;
    const int wave  = tid >> 5;
    const int waveM = wave >> 2;      // 0..1 -> 64-row slab
    const int waveN = wave & 3;       // 0..3 -> 32-col slab
    const int rl    = lane & 15;      // row/col within a 16-wide fragment
    const int hi    = lane >> 4;      // selects K pair {0,1} vs {2,3}

    const int n0 = blockIdx.x * BN;
    const int m0 = blockIdx.y * BM;

    // gather one expert's weight matrix (uniform scalar load)
    const float* __restrict__ wb =
        w + (size_t)(*expert_id) * (size_t)OUT_DIM * (size_t)IN_DIM;

    // each thread DMAs 2 x b128 from the A-tile and 2 from the B-tile per K step
    const int q  = tid & 3;           // which float4 of the 16-wide K tile
    const int r0 = tid >> 2;          // 0..63
    const float* aPtr0 = state + (size_t)(m0 + r0)      * IN_DIM + q * 4;
    const float* aPtr1 = state + (size_t)(m0 + r0 + 64) * IN_DIM + q * 4;
    const float* bPtr0 = wb    + (size_t)(n0 + r0)      * IN_DIM + q * 4;
    const float* bPtr1 = wb    + (size_t)(n0 + r0 + 64) * IN_DIM + q * 4;

    const int ldsA0 = r0        * LDS_STRIDE + q * 4;
    const int ldsA1 = (r0 + 64) * LDS_STRIDE + q * 4;

    v8f acc[4][2];
#pragma unroll
    for (int i = 0; i < 4; ++i)
#pragma unroll
        for (int j = 0; j < 2; ++j)
            acc[i][j] = (v8f){};

    const int K_TILES = IN_DIM / BK;  // 256

    // prologue: kick off tile 0 into buffer 0
    {
        async_cp_b128(aPtr0, &As[0][ldsA0]);
        async_cp_b128(aPtr1, &As[0][ldsA1]);
        async_cp_b128(bPtr0, &Bs[0][ldsA0]);
        async_cp_b128(bPtr1, &Bs[0][ldsA1]);
    }

    for (int kt = 0; kt < K_TILES; ++kt) {
        const int cur = kt & 1;
        const int nxt = cur ^ 1;

        // all waves finished computing from buffer `nxt` (used in the previous iteration)
        __syncthreads();

        if (kt + 1 < K_TILES) {
            const size_t koff = (size_t)(kt + 1) * BK;
            async_cp_b128(aPtr0 + koff, &As[nxt][ldsA0]);
            async_cp_b128(aPtr1 + koff, &As[nxt][ldsA1]);
            async_cp_b128(bPtr0 + koff, &Bs[nxt][ldsA0]);
            async_cp_b128(bPtr1 + koff, &Bs[nxt][ldsA1]);
            wait_async<4>();          // in-order: the 4 loads of tile `kt` are done
        } else {
            wait_async<0>();
        }
        __syncthreads();              // tile `kt` visible to every wave

#pragma unroll
        for (int kk = 0; kk < BK; kk += 4) {
            v2f afrag[4], bfrag[2];
#pragma unroll
            for (int i = 0; i < 4; ++i) {
                const int m = waveM * 64 + i * 16 + rl;
                afrag[i] = *(const v2f*)&As[cur][m * LDS_STRIDE + kk + 2 * hi];
            }
#pragma unroll
            for (int j = 0; j < 2; ++j) {
                const int n = waveN * 32 + j * 16 + rl;
                bfrag[j] = *(const v2f*)&Bs[cur][n * LDS_STRIDE + kk + 2 * hi];
            }
#pragma unroll
            for (int i = 0; i < 4; ++i)
#pragma unroll
                for (int j = 0; j < 2; ++j)
                    acc[i][j] = __builtin_amdgcn_wmma_f32_16x16x4_f32(
                        /*neg_a=*/false, afrag[i],
                        /*neg_b=*/false, bfrag[j],
                        /*c_mod=*/(short)0, acc[i][j],
                        /*reuse_a=*/false, /*reuse_b=*/false);
        }
    }

    // epilogue: C/D layout -> lane gives N, VGPR index v gives M = v + 8*hi
#pragma unroll
    for (int i = 0; i < 4; ++i) {
        const int mBase = m0 + waveM * 64 + i * 16 + hi * 8;
#pragma unroll
        for (int j = 0; j < 2; ++j) {
            const int n = n0 + waveN * 32 + j * 16 + rl;
#pragma unroll
            for (int v = 0; v < 8; ++v)
                out[(size_t)(mBase + v) * OUT_DIM + n] = acc[i][j][v];
        }
    }
}

extern "C" void kernel_launch(void* const* d_in, const int* in_sizes, int n_in,
                              void* d_out, int out_size, void* d_ws, size_t ws_size,
                              hipStream_t stream) {
    (void)in_sizes; (void)n_in; (void)d_ws; (void)ws_size; (void)out_size;
    const float* state     = (const float*)d_in[0];
    const float* w         = (const float*)d_in[1];
    const int*   expert_id = (const int*)d_in[2];
    float*       out       = (float*)d_out;

    dim3 grid(OUT_DIM / BN, TOKENS / BM);   // 112 x 32
    dim3 block(256);
    moe_gemm_f32_wmma<<<grid, block, 0, stream>>>(state, w, expert_id, out);
}